// AttentionAggregation_61649960566785
// MI455X (gfx1250) — compile-verified
//
#include <hip/hip_runtime.h>

typedef __attribute__((ext_vector_type(2))) float v2f;
typedef __attribute__((ext_vector_type(8))) float v8f;

#define FD 128      // feature dim
#define TILE 16     // edges per wave

// ---------------- zero accumulators ----------------
__global__ void zero_kernel(float* __restrict__ out, int n_out,
                            float* __restrict__ cnt, int n_cnt) {
  int i = blockIdx.x * blockDim.x + threadIdx.x;
  if (i < n_out) out[i] = 0.0f;
  if (i < n_cnt) cnt[i] = 0.0f;
}

// ---------------- main edge kernel: WMMA logits + atomic scatter ----------------
__global__ __launch_bounds__(256) void edge_attn_kernel(
    const float* __restrict__ src_feat,      // [n_src, FD]
    const float* __restrict__ dst_feat,      // [n_dst, FD]
    const float* __restrict__ att_w,         // [2*FD]
    const float* __restrict__ att_b,         // [1]
    const long long* __restrict__ edge_index,// [2, E]
    float* __restrict__ agg,                 // [n_dst, FD]  (d_out, accumulated)
    float* __restrict__ cnt,                 // [n_dst]      (d_ws)
    int E, int ntiles)
{
  __shared__ float w_lds[2 * FD];
  int tid = threadIdx.x;
  if (tid < 2 * FD) w_lds[tid] = att_w[tid];
  __syncthreads();

  int lane = tid & 31;
  int tile = blockIdx.x * (blockDim.x >> 5) + (tid >> 5);
  if (tile >= ntiles) return;                 // uniform per-wave exit

  int base = tile * TILE;
  int rem  = E - base; if (rem > TILE) rem = TILE;   // uniform per wave

  // Edge handled by this lane: lane n and lane n+16 both carry edge (n&15).
  int eo = lane & 15;
  int ce = (eo < rem) ? eo : (rem - 1);       // clamp for partial tiles
  long long sidx = edge_index[base + ce];
  long long didx = edge_index[E + base + ce];

  const float* srow = src_feat + sidx * FD;
  const float* drow = dst_feat + didx * FD;
  // B (4x16, K x N) f32 layout mirrors documented A layout:
  // lanes 0-15 hold K=0,1 ; lanes 16-31 hold K=2,3 -> one aligned float2 load.
  int koff = (lane < 16) ? 0 : 2;

  v8f acc = {};
  #pragma unroll 4
  for (int c = 0; c < FD / 4; ++c) {
    int k = 4 * c + koff;
    v2f bs = *(const v2f*)(srow + k);          // B chunk: src rows, edges in N
    v2f bd = *(const v2f*)(drow + k);          // B chunk: dst rows
    v2f as = *(const v2f*)(&w_lds[k]);         // A chunk: w_src broadcast over M
    v2f ad = *(const v2f*)(&w_lds[FD + k]);    // A chunk: w_dst broadcast over M
    // logits[n] += sum_k w_src[k]*src[n,k]  (every D row identical)
    acc = __builtin_amdgcn_wmma_f32_16x16x4_f32(false, as, false, bs,
                                                (short)0, acc, false, false);
    acc = __builtin_amdgcn_wmma_f32_16x16x4_f32(false, ad, false, bd,
                                                (short)0, acc, false, false);
  }

  // D[m][n] identical over m; lane n holds logit of edge (n&15).
  float logit = acc[0] + att_b[0];
  float att = 1.0f / (1.0f + __expf(-logit));  // sigmoid

  // Scatter: wave cooperates on one edge row at a time (float4 per lane).
  for (int e = 0; e < rem; ++e) {
    float a_e      = __shfl(att, e, 32);
    long long s_e  = __shfl(sidx, e, 32);
    long long d_e  = __shfl(didx, e, 32);
    const float4 v = *(const float4*)(src_feat + s_e * FD + lane * 4);
    float* orow = agg + d_e * FD + lane * 4;
    unsafeAtomicAdd(orow + 0, v.x * a_e);
    unsafeAtomicAdd(orow + 1, v.y * a_e);
    unsafeAtomicAdd(orow + 2, v.z * a_e);
    unsafeAtomicAdd(orow + 3, v.w * a_e);
    if (lane == 0) unsafeAtomicAdd(cnt + d_e, a_e);
  }
}

// ---------------- finalize: divide by clamped count ----------------
__global__ void div_kernel(float* __restrict__ out, const float* __restrict__ cnt,
                           int n_dst) {
  int i = blockIdx.x * blockDim.x + threadIdx.x;   // one float4 per thread
  int total = n_dst * (FD / 4);
  if (i >= total) return;
  int row = i / (FD / 4);
  float c = fmaxf(cnt[row], 1e-8f);
  float inv = 1.0f / c;
  float4* p = (float4*)out + i;
  float4 v = *p;
  v.x *= inv; v.y *= inv; v.z *= inv; v.w *= inv;
  *p = v;
}

extern "C" void kernel_launch(void* const* d_in, const int* in_sizes, int n_in,
                              void* d_out, int out_size, void* d_ws, size_t ws_size,
                              hipStream_t stream) {
  const float* src_feat = (const float*)d_in[0];
  const float* dst_feat = (const float*)d_in[1];
  const float* att_w    = (const float*)d_in[2];
  const float* att_b    = (const float*)d_in[3];
  const long long* edge_index = (const long long*)d_in[4];

  int E     = in_sizes[4] / 2;       // edge_index is [2, E]
  int d     = in_sizes[2] / 2;       // att_w is [2D] -> D == 128
  int n_dst = out_size / d;

  float* out = (float*)d_out;
  float* cnt = (float*)d_ws;         // n_dst floats of scratch

  // 1) zero accumulators (agg lives in d_out, count in d_ws)
  zero_kernel<<<(out_size + 255) / 256, 256, 0, stream>>>(out, out_size, cnt, n_dst);

  // 2) edge attention + scatter: 1 wave = 16 edges, 8 waves per block
  int ntiles = (E + TILE - 1) / TILE;
  int blocks = (ntiles + 7) / 8;
  edge_attn_kernel<<<blocks, 256, 0, stream>>>(src_feat, dst_feat, att_w, att_b,
                                               edge_index, out, cnt, E, ntiles);

  // 3) normalize
  int total4 = n_dst * (d / 4);
  div_kernel<<<(total4 + 255) / 256, 256, 0, stream>>>(out, cnt, n_dst);
}